// MSSSIM_3135326126171
// MI455X (gfx1250) — compile-verified
//
#include <hip/hip_runtime.h>

// MS-SSIM 3D for MI455X (gfx1250). All separable 11-tap Gaussian conv passes run on the
// matrix pipe as banded-matrix GEMMs using V_WMMA_F32_16X16X4_F32 (fp32 in/out, wave32).

typedef float v2f __attribute__((ext_vector_type(2)));
typedef float v8f __attribute__((ext_vector_type(8)));

static __device__ __forceinline__ v8f wmma4(v2f a, v2f b, v8f c) {
    // D(16x16) += A(16x4) * B(4x16), fp32.
    return __builtin_amdgcn_wmma_f32_16x16x4_f32(false, a, false, b, (short)0, c, false, false);
}

// Banded weight matrix element: B[k][n] = g[k-n] for 0<=k-n<=10 else 0.
// Computed analytically (sigma=1.5, normalized): g(t) = inv * exp2(-(t-5)^2 / (2*sigma^2) * log2e)
// Branchless: v_exp_f32 + v_cndmask, no memory traffic, no EXEC changes.
static __device__ __forceinline__ float bandW(int k, int n) {
    float d = (float)(k - n - 5);
    float g = 0.26601172f * __builtin_amdgcn_exp2f(-0.32059890f * d * d);
    return (__builtin_fabsf(d) <= 5.0f) ? g : 0.0f;
}

// ---------------------------------------------------------------------------
// Pass 1: X-axis conv + on-the-fly products. One wave = 16 rows x 16 w-outputs.
// rows r = zz*H + h over a 26-slice z slab. Writes 5 fields to tmpA.
// A lane layout (16x4 fp32): lane = M%16, VGPR r holds K = 2*(lane/16)+r  -> contiguous
// K-pair per lane => single b64 load per image per chunk on the interior path.
// ---------------------------------------------------------------------------
__global__ void k_convx(const float* __restrict__ i1, const float* __restrict__ i2,
                        float* __restrict__ tA,
                        int n, int S, int z0, int wT, int totalWaves)
{
    int gid = (int)blockIdx.x * 8 + ((int)threadIdx.x >> 5);
    if (gid >= totalWaves) return;               // wave-uniform, EXEC stays all-ones
    int lane = (int)threadIdx.x & 31;
    int m = lane & 15, half = lane >> 4;
    int wt = gid % wT, rt = gid / wT;
    int H = S, Wo = S - 10;
    int rows = 26 * H;
    int r = rt * 16 + m; if (r > rows - 1) r = rows - 1;   // clamp (masked at store)
    int zz = r / H, h = r - zz * H;
    int zin = z0 + zz; if (zin > S - 1) zin = S - 1;
    size_t rowbase = (((size_t)n * S + zin) * S + h) * (size_t)S;
    int w0 = wt * 16;

    v8f a0 = {}, a1v = {}, a2v = {}, a3v = {}, a4v = {};
    if (w0 + 27 < S) {
        // Interior tile (wave-uniform): contiguous, 8B-aligned v2f loads.
        const float* p1 = i1 + rowbase + w0 + 2 * half;
        const float* p2 = i2 + rowbase + w0 + 2 * half;
        for (int ch = 0; ch < 7; ++ch) {
            int kg = 4 * ch + 2 * half;
            v2f x1 = *(const v2f*)(p1 + 4 * ch);
            v2f x2 = *(const v2f*)(p2 + 4 * ch);
            v2f b; b.x = bandW(kg, m); b.y = bandW(kg + 1, m);
            v2f p11 = x1 * x1, p22 = x2 * x2, p12 = x1 * x2;
            a0  = wmma4(x1,  b, a0);
            a1v = wmma4(x2,  b, a1v);
            a2v = wmma4(p11, b, a2v);
            a3v = wmma4(p22, b, a3v);
            a4v = wmma4(p12, b, a4v);
        }
    } else {
        // Edge tile: clamped taps only feed zero weights for valid outputs.
        for (int ch = 0; ch < 7; ++ch) {
            int kg = 4 * ch + 2 * half;
            int wa = w0 + kg;     if (wa > S - 1) wa = S - 1;
            int wb = w0 + kg + 1; if (wb > S - 1) wb = S - 1;
            v2f x1, x2;
            x1.x = i1[rowbase + wa]; x1.y = i1[rowbase + wb];
            x2.x = i2[rowbase + wa]; x2.y = i2[rowbase + wb];
            v2f b; b.x = bandW(kg, m); b.y = bandW(kg + 1, m);
            v2f p11 = x1 * x1, p22 = x2 * x2, p12 = x1 * x2;
            a0  = wmma4(x1,  b, a0);
            a1v = wmma4(x2,  b, a1v);
            a2v = wmma4(p11, b, a2v);
            a3v = wmma4(p22, b, a3v);
            a4v = wmma4(p12, b, a4v);
        }
    }
    size_t plane = (size_t)26 * H * Wo;
    int wcol = w0 + m;                            // D: col N = lane%16
    bool wok = wcol < Wo;
    for (int v = 0; v < 8; ++v) {
        int rr = rt * 16 + v + 8 * half;          // D: row M = v + 8*(lane/16)
        if (wok && rr < rows) {
            size_t o = (size_t)rr * Wo + wcol;
            tA[o]             = a0[v];
            tA[plane + o]     = a1v[v];
            tA[2 * plane + o] = a2v[v];
            tA[3 * plane + o] = a3v[v];
            tA[4 * plane + o] = a4v[v];
        }
    }
}

// ---------------------------------------------------------------------------
// Pass 2: Y-axis conv per field. M = 16 w-positions, N = 16 h'-outputs, K = 28 h-taps.
// ---------------------------------------------------------------------------
__global__ void k_convy(const float* __restrict__ tA, float* __restrict__ tB,
                        int S, int hT, int wT, int totalWaves)
{
    int gid = (int)blockIdx.x * 8 + ((int)threadIdx.x >> 5);
    if (gid >= totalWaves) return;
    int lane = (int)threadIdx.x & 31;
    int m = lane & 15, half = lane >> 4;
    int wt = gid % wT; int q = gid / wT;
    int ht = q % hT; q /= hT;
    int zz = q % 26; int f = q / 26;
    int H = S, Ho = S - 10, Wo = S - 10;
    size_t planeA = (size_t)26 * H * Wo;
    size_t planeB = (size_t)26 * Ho * Wo;
    const float* src = tA + (size_t)f * planeA + (size_t)zz * H * Wo;
    int h0 = ht * 16, w0 = wt * 16;
    int wm = w0 + m; if (wm > Wo - 1) wm = Wo - 1;

    v8f acc = {};
    for (int ch = 0; ch < 7; ++ch) {
        int kg = 4 * ch + 2 * half;
        int ha = h0 + kg;     if (ha > H - 1) ha = H - 1;
        int hb = h0 + kg + 1; if (hb > H - 1) hb = H - 1;
        v2f a; a.x = src[(size_t)ha * Wo + wm]; a.y = src[(size_t)hb * Wo + wm];
        v2f b; b.x = bandW(kg, m); b.y = bandW(kg + 1, m);
        acc = wmma4(a, b, acc);
    }
    float* dst = tB + (size_t)f * planeB + (size_t)zz * Ho * Wo;
    int hcol = h0 + m;
    bool hok = hcol < Ho;
    for (int v = 0; v < 8; ++v) {
        int ww = w0 + v + 8 * half;
        if (hok && ww < Wo) dst[(size_t)hcol * Wo + ww] = acc[v];
    }
}

// ---------------------------------------------------------------------------
// Pass 3: Z-axis conv (all 5 fields in one wave) + SSIM map + reduction.
// M = 16 w-positions, N = 16 z'-outputs, K = 26 slab slices (pad -> zero weights).
// ---------------------------------------------------------------------------
__global__ void k_ssimz(const float* __restrict__ tB,
                        double* __restrict__ acc2, const int* __restrict__ flg,
                        int S, int z0, int wT, int totalWaves)
{
    int gid = (int)blockIdx.x * 8 + ((int)threadIdx.x >> 5);
    if (gid >= totalWaves) return;
    int lane = (int)threadIdx.x & 31;
    int m = lane & 15, half = lane >> 4;
    int wt = gid % wT; int hp = gid / wT;
    int Ho = S - 10, Wo = S - 10, Do = S - 10;
    size_t sliceB = (size_t)Ho * Wo;
    size_t planeB = (size_t)26 * sliceB;
    size_t rowoff = (size_t)hp * Wo;
    int w0 = wt * 16;
    int wm = w0 + m; if (wm > Wo - 1) wm = Wo - 1;

    v8f c0 = {}, c1 = {}, c2 = {}, c3 = {}, c4 = {};
    for (int ch = 0; ch < 7; ++ch) {
        int kg = 4 * ch + 2 * half;
        int za = kg;     if (za > 25) za = 25;   // kg>=26 has zero weight for all valid N
        int zb = kg + 1; if (zb > 25) zb = 25;
        v2f b; b.x = bandW(kg, m); b.y = bandW(kg + 1, m);
        size_t oa = (size_t)za * sliceB + rowoff + wm;
        size_t ob = (size_t)zb * sliceB + rowoff + wm;
        v2f a;
        a.x = tB[oa];              a.y = tB[ob];              c0 = wmma4(a, b, c0);
        a.x = tB[planeB + oa];     a.y = tB[planeB + ob];     c1 = wmma4(a, b, c1);
        a.x = tB[2 * planeB + oa]; a.y = tB[2 * planeB + ob]; c2 = wmma4(a, b, c2);
        a.x = tB[3 * planeB + oa]; a.y = tB[3 * planeB + ob]; c3 = wmma4(a, b, c3);
        a.x = tB[4 * planeB + oa]; a.y = tB[4 * planeB + ob]; c4 = wmma4(a, b, c4);
    }

    float L = (flg[0] ? 255.0f : 1.0f) + (flg[1] ? 1.0f : 0.0f);
    float C1 = (0.01f * L) * (0.01f * L);
    float C2v = (0.03f * L) * (0.03f * L);
    float ssum = 0.0f, csum = 0.0f;
    int zq = z0 + m;                              // D col N = z'-in-slab
    for (int v = 0; v < 8; ++v) {
        int wq = w0 + v + 8 * half;               // D row M = w'
        if (wq < Wo && zq < Do) {
            float mu1 = c0[v], mu2 = c1[v];
            float s1 = c2[v] - mu1 * mu1;
            float s2 = c3[v] - mu2 * mu2;
            float s12 = c4[v] - mu1 * mu2;
            float v1 = 2.0f * s12 + C2v;
            float v2 = s1 + s2 + C2v;
            csum += v1 / v2;
            ssum += ((2.0f * mu1 * mu2 + C1) * v1) / ((mu1 * mu1 + mu2 * mu2 + C1) * v2);
        }
    }
    for (int off = 16; off > 0; off >>= 1) {
        ssum += __shfl_down(ssum, off, 32);
        csum += __shfl_down(csum, off, 32);
    }
    if (lane == 0) {
        atomicAdd(&acc2[0], (double)ssum);        // GLOBAL_ATOMIC_ADD_F64
        atomicAdd(&acc2[1], (double)csum);
    }
}

// 2x2x2 average pool, one image (2,1,S,S,S) -> (2,1,S/2,S/2,S/2)
__global__ void k_pool(const float* __restrict__ src, float* __restrict__ dst, int S)
{
    int So = S >> 1;
    size_t tot = (size_t)2 * So * So * So;
    for (size_t i = (size_t)blockIdx.x * blockDim.x + threadIdx.x; i < tot;
         i += (size_t)gridDim.x * blockDim.x) {
        size_t q = i;
        int x = (int)(q % So); q /= So;
        int y = (int)(q % So); q /= So;
        int z = (int)(q % So); int n = (int)(q / So);
        float s = 0.0f;
        for (int dz = 0; dz < 2; ++dz)
            for (int dy = 0; dy < 2; ++dy)
                for (int dx = 0; dx < 2; ++dx) {
                    size_t o = ((((size_t)n * S + (2 * z + dz)) * S + (2 * y + dy)) * S
                                + (2 * x + dx));
                    s += src[o];
                }
        dst[i] = s * 0.125f;
    }
}

// Per-level range heuristic flags: flg[0] = any(v>128), flg[1] = any(v<-0.5)
__global__ void k_flags(const float* __restrict__ img, unsigned long long nElem,
                        int* __restrict__ flg)
{
    bool pmax = false, pmin = false;
    for (size_t i = (size_t)blockIdx.x * blockDim.x + threadIdx.x; i < nElem;
         i += (size_t)gridDim.x * blockDim.x) {
        float v = img[i];
        pmax |= (v > 128.0f);
        pmin |= (v < -0.5f);
    }
    int lane = (int)threadIdx.x & 31;
    if (__ballot(pmax)) { if (lane == 0) atomicOr(&flg[0], 1); }
    if (__ballot(pmin)) { if (lane == 0) atomicOr(&flg[1], 1); }
}

__global__ void k_init(double* acc, int* flg)
{
    int i = (int)threadIdx.x;
    if (i < 10) { acc[i] = 0.0; flg[i] = 0; }
}

__global__ void k_final(const double* __restrict__ acc, float* __restrict__ out)
{
    if (threadIdx.x == 0 && blockIdx.x == 0) {
        const double w[5] = {0.0448, 0.2856, 0.3001, 0.2363, 0.1333};
        double res = 1.0;
        for (int l = 0; l < 5; ++l) {
            int S = 192 >> l, Do = S - 10;
            double cnt = 2.0 * (double)Do * (double)Do * (double)Do;
            double sm = acc[2 * l] / cnt;
            double cm = acc[2 * l + 1] / cnt;
            res *= (l < 4) ? pow(cm, w[l]) : pow(sm, w[4]);
        }
        out[0] = (float)res;
    }
}

extern "C" void kernel_launch(void* const* d_in, const int* in_sizes, int n_in,
                              void* d_out, int out_size, void* d_ws, size_t ws_size,
                              hipStream_t stream)
{
    (void)in_sizes; (void)n_in; (void)out_size; (void)ws_size;
    const float* img1_0 = (const float*)d_in[0];
    const float* img2_0 = (const float*)d_in[1];
    float* out = (float*)d_out;

    char* ws = (char*)d_ws;
    size_t cur = 0;
    auto alloc = [&](size_t bytes) {
        size_t o = cur;
        cur = (cur + bytes + 255) & ~(size_t)255;
        return o;
    };

    double* acc = (double*)(ws + alloc(10 * sizeof(double)));
    int*    flg = (int*)(ws + alloc(10 * sizeof(int)));
    float*  tA  = (float*)(ws + alloc((size_t)5 * 26 * 192 * 182 * sizeof(float)));
    float*  tB  = (float*)(ws + alloc((size_t)5 * 26 * 182 * 182 * sizeof(float)));

    const float* i1p[5]; const float* i2p[5];
    float* i1s[5] = {}; float* i2s[5] = {};
    i1p[0] = img1_0; i2p[0] = img2_0;
    for (int l = 1; l < 5; ++l) {
        int S = 192 >> l;
        size_t e = (size_t)2 * S * S * S * sizeof(float);
        i1s[l] = (float*)(ws + alloc(e));
        i2s[l] = (float*)(ws + alloc(e));
        i1p[l] = i1s[l]; i2p[l] = i2s[l];
    }

    k_init<<<1, 64, 0, stream>>>(acc, flg);

    for (int l = 0; l < 5; ++l) {
        int S = 192 >> l;
        int Do = S - 10, Ho = S - 10, Wo = S - 10;

        unsigned long long ne = (unsigned long long)2 * S * S * S;
        int fb = (int)((ne + 255) / 256); if (fb > 2048) fb = 2048;
        k_flags<<<fb, 256, 0, stream>>>(i1p[l], ne, flg + 2 * l);

        int wT = (Wo + 15) / 16;
        int hT = (Ho + 15) / 16;
        int rT = (26 * S + 15) / 16;
        int nslab = (Do + 15) / 16;
        for (int n = 0; n < 2; ++n) {
            for (int sb = 0; sb < nslab; ++sb) {
                int z0 = sb * 16;
                int t1 = rT * wT;
                k_convx<<<(t1 + 7) / 8, 256, 0, stream>>>(i1p[l], i2p[l], tA, n, S, z0, wT, t1);
                int t2 = 5 * 26 * hT * wT;
                k_convy<<<(t2 + 7) / 8, 256, 0, stream>>>(tA, tB, S, hT, wT, t2);
                int t3 = Ho * wT;
                k_ssimz<<<(t3 + 7) / 8, 256, 0, stream>>>(tB, acc + 2 * l, flg + 2 * l,
                                                          S, z0, wT, t3);
            }
        }
        if (l < 4) {
            int So = S >> 1;
            size_t tot = (size_t)2 * So * So * So;
            int pb = (int)((tot + 255) / 256); if (pb > 4096) pb = 4096;
            k_pool<<<pb, 256, 0, stream>>>(i1p[l], i1s[l + 1], S);
            k_pool<<<pb, 256, 0, stream>>>(i2p[l], i2s[l + 1], S);
        }
    }
    k_final<<<1, 1, 0, stream>>>(acc, out);
}